// GAT_VS_42125039239515
// MI455X (gfx1250) — compile-verified
//
#include <hip/hip_runtime.h>
#include <hip/hip_bf16.h>

typedef __attribute__((ext_vector_type(2))) float v2f;
typedef __attribute__((ext_vector_type(8))) float v8f;

#define NEG_SLOPE 0.2f

// ---- monotone float <-> ordered-uint encoding for atomicMax on floats ----
__device__ __forceinline__ unsigned f2ord(float f) {
  unsigned u = __float_as_uint(f);
  return (u & 0x80000000u) ? ~u : (u | 0x80000000u);
}
__device__ __forceinline__ float ord2f(unsigned u) {
  return (u & 0x80000000u) ? __uint_as_float(u & 0x7FFFFFFFu)
                           : __uint_as_float(~u);
}

// native fp32 atomic add (no return -> STOREcnt; S_ENDPGM wait-idle completes it)
__device__ __forceinline__ void atomic_add_f32(float* p, float v) {
  asm volatile("global_atomic_add_f32 %0, %1, off" :: "v"(p), "v"(v) : "memory");
}

// Repack W[K,C] into float2 pairs Wp[k/2][C] = {W[k][c], W[k+1][c]}
// so a WMMA B-fragment is a single b64 load.
__global__ void gat_repack_w(const float* __restrict__ W,
                             float2* __restrict__ Wp, int pairs, int C) {
  int i = blockIdx.x * blockDim.x + threadIdx.x;  // over (K/2)*C
  if (i >= pairs) return;
  int kp = i / C, c = i - kp * C;
  Wp[i] = make_float2(W[(2 * kp) * C + c], W[(2 * kp + 1) * C + c]);
}

// =====================================================================
// GEMM: H[N,COUT] = A[N,K] * W[K,COUT]  (exact fp32, WMMA 16x16x4)
// One wave -> 16-row x COUT strip. Software-pipelined: next K-step's
// 1 A-fragment + NT B-fragments are issued before the current WMMAs.
// =====================================================================
template <int COUT>
__global__ void gat_gemm_wmma(const float* __restrict__ A,
                              const float2* __restrict__ Wp,
                              float* __restrict__ H, int N, int K) {
  constexpr int NT = COUT / 16;
  const int lane = threadIdx.x & 31;
  const int wave = threadIdx.x >> 5;
  const int row0 = (blockIdx.x * 4 + wave) * 16;
  if (row0 >= N) return;  // wave-uniform -> EXEC all-1s at WMMA

  const int half = lane >> 4;   // 0: K pair {0,1}, 1: K pair {2,3}
  const int r    = lane & 15;
  int arow = row0 + r;
  if (arow >= N) arow = N - 1;  // clamp loads; stores guarded below
  const v2f* arow2 = (const v2f*)(A + (size_t)arow * K);
  const v2f* wl    = (const v2f*)Wp + r;   // lane's column base

  v8f acc[NT];
  const v8f vz = {0.f, 0.f, 0.f, 0.f, 0.f, 0.f, 0.f, 0.f};
#pragma unroll
  for (int t = 0; t < NT; ++t) acc[t] = vz;

  int kp = half;                         // float2-pair index for this lane
  v2f a_cur = arow2[kp];
  v2f b_cur[NT];
#pragma unroll
  for (int t = 0; t < NT; ++t) b_cur[t] = wl[kp * COUT + t * 16];

  const int steps = K >> 2;
  for (int s = 0; s < steps; ++s) {
    const int kpn = (s + 1 < steps) ? (kp + 2) : kp;  // clamp on last iter
    // ---- prefetch next K-step's fragments (in flight during WMMAs) ----
    v2f a_nxt = arow2[kpn];
    v2f b_nxt[NT];
#pragma unroll
    for (int t = 0; t < NT; ++t) b_nxt[t] = wl[kpn * COUT + t * 16];
    // ---- matrix math on current fragments ----
#pragma unroll
    for (int t = 0; t < NT; ++t)
      acc[t] = __builtin_amdgcn_wmma_f32_16x16x4_f32(
          false, a_cur, false, b_cur[t], (short)0, acc[t], false, false);
    a_cur = a_nxt;
#pragma unroll
    for (int t = 0; t < NT; ++t) b_cur[t] = b_nxt[t];
    kp = kpn;
  }

  // C/D layout: VGPR j -> row row0 + j + half*8, col = t*16 + r
  const int orow0 = row0 + half * 8;
#pragma unroll
  for (int t = 0; t < NT; ++t) {
    const int col = t * 16 + r;
#pragma unroll
    for (int j = 0; j < 8; ++j) {
      const int rr = orow0 + j;
      if (rr < N) H[(size_t)rr * COUT + col] = acc[t][j];
    }
  }
}

// e_src[n] = h[n].a_src ; e_dst[n] = h[n].a_dst
__global__ void gat_attn_coef(const float* __restrict__ H,
                              const float* __restrict__ avs,
                              const float* __restrict__ avd,
                              float* __restrict__ es, float* __restrict__ ed,
                              int N, int C) {
  int n = blockIdx.x * blockDim.x + threadIdx.x;
  if (n >= N) return;
  const float4* h  = (const float4*)(H + (size_t)n * C);
  const float4* s4 = (const float4*)avs;
  const float4* d4 = (const float4*)avd;
  float s = 0.f, d = 0.f;
  for (int c = 0; c < (C >> 2); ++c) {
    float4 hv = h[c], sv = s4[c], dv = d4[c];
    s += hv.x * sv.x + hv.y * sv.y + hv.z * sv.z + hv.w * sv.w;
    d += hv.x * dv.x + hv.y * dv.y + hv.z * dv.z + hv.w * dv.w;
  }
  es[n] = s;
  ed[n] = d;
}

// zero aggregation buffer; init segment-max (encode(-inf)) and denom
__global__ void gat_init(unsigned* __restrict__ mord, float* __restrict__ denom,
                         float* __restrict__ agg, int N, long total) {
  long i = blockIdx.x * (long)blockDim.x + threadIdx.x;
  if (i < total) agg[i] = 0.f;
  if (i < N) { mord[i] = 0x007FFFFFu; denom[i] = 0.f; }  // f2ord(-inf)
}

// edge -> (src, dst), self-loops appended virtually
__device__ __forceinline__ void edge_sd(const int* __restrict__ ei, int E,
                                        int e, int& s, int& d) {
  if (e < E) { s = ei[e]; d = ei[E + e]; }
  else       { s = e - E; d = s; }
}

__global__ void gat_edge_logit_max(const int* __restrict__ ei, int E, int N,
                                   const float* __restrict__ es,
                                   const float* __restrict__ ed,
                                   float* __restrict__ elog,
                                   unsigned* __restrict__ mord) {
  int e = blockIdx.x * blockDim.x + threadIdx.x;
  if (e >= E + N) return;
  int s, d;
  edge_sd(ei, E, e, s, d);
  float l = es[s] + ed[d];
  l = (l > 0.f) ? l : NEG_SLOPE * l;
  elog[e] = l;
  atomicMax(mord + d, f2ord(l));
}

__global__ void gat_edge_exp_sum(const int* __restrict__ ei, int E, int N,
                                 float* __restrict__ elog,
                                 const unsigned* __restrict__ mord,
                                 float* __restrict__ denom) {
  int e = blockIdx.x * blockDim.x + threadIdx.x;
  if (e >= E + N) return;
  int s, d;
  edge_sd(ei, E, e, s, d);
  float ee = __expf(elog[e] - ord2f(mord[d]));
  elog[e] = ee;  // overwrite logit with exp value
  atomic_add_f32(denom + d, ee);
}

// one wave32 per edge: agg[dst] += alpha * h[src]
// float4 row gather (global_load_b128, 512B coalesced) + native f32 atomics
__global__ void gat_edge_aggregate(const int* __restrict__ ei, int E, int N,
                                   int C4, const float* __restrict__ eexp,
                                   const float* __restrict__ denom,
                                   const float* __restrict__ H,
                                   float* __restrict__ agg) {
  int w = (int)((blockIdx.x * (long)blockDim.x + threadIdx.x) >> 5);
  int lane = threadIdx.x & 31;
  if (w >= E + N) return;
  int s, d;
  edge_sd(ei, E, w, s, d);
  float alpha = eexp[w] / denom[d];
  const float4* hs = (const float4*)(H + (size_t)s * (C4 * 4));
  float* od = agg + (size_t)d * (C4 * 4);
  for (int c4 = lane; c4 < C4; c4 += 32) {
    float4 hv = hs[c4];
    float* p = od + c4 * 4;
    atomic_add_f32(p + 0, alpha * hv.x);
    atomic_add_f32(p + 1, alpha * hv.y);
    atomic_add_f32(p + 2, alpha * hv.z);
    atomic_add_f32(p + 3, alpha * hv.w);
  }
}

__global__ void gat_bias_act(const float* __restrict__ agg,
                             const float* __restrict__ b,
                             float* __restrict__ out, long total, int cmask,
                             int relu) {
  long i = blockIdx.x * (long)blockDim.x + threadIdx.x;
  if (i >= total) return;
  float v = agg[i] + b[i & cmask];
  if (relu) v = fmaxf(v, 0.f);
  out[i] = v;
}

extern "C" void kernel_launch(void* const* d_in, const int* in_sizes, int n_in,
                              void* d_out, int out_size, void* d_ws,
                              size_t ws_size, hipStream_t stream) {
  const float* x   = (const float*)d_in[0];
  const int*   ei  = (const int*)d_in[1];
  const float* W0  = (const float*)d_in[2];
  const float* as0 = (const float*)d_in[3];
  const float* ad0 = (const float*)d_in[4];
  const float* b0  = (const float*)d_in[5];
  const float* W1  = (const float*)d_in[6];
  const float* as1 = (const float*)d_in[7];
  const float* ad1 = (const float*)d_in[8];
  const float* b1  = (const float*)d_in[9];
  const float* W2  = (const float*)d_in[10];
  const float* as2 = (const float*)d_in[11];
  const float* ad2 = (const float*)d_in[12];
  const float* b2  = (const float*)d_in[13];

  const int CHID = in_sizes[3];         // 128
  const int CIN  = in_sizes[2] / CHID;  // 128
  const int COUT = in_sizes[11];        // 64
  const int N    = in_sizes[0] / CIN;   // 50000
  const int E    = in_sizes[1] / 2;     // 1.6M
  const int ET   = E + N;               // with self loops

  float* ws      = (float*)d_ws;
  float* h       = ws;              ws += (size_t)N * CHID;
  float* agg     = ws;              ws += (size_t)N * CHID;
  float* act     = ws;              ws += (size_t)N * CHID;
  float* es      = ws;              ws += N;
  float* ed      = ws;              ws += N;
  unsigned* mord = (unsigned*)ws;   ws += N;
  float* denom   = ws;              ws += N;
  float* elog    = ws;              ws += (ET + 1) & ~1;  // keep next 8B-aligned
  float2* wpack  = (float2*)ws;     // (K/2)*C pairs, max 64KB

  auto layer = [&](const float* X, const float* Wm, const float* avs,
                   const float* avd, const float* bv, int K, int C, float* O,
                   bool relu) {
    const int pairs = (K / 2) * C;
    gat_repack_w<<<(pairs + 255) / 256, 256, 0, stream>>>(Wm, wpack, pairs, C);

    dim3 gblk(128);                       // 4 waves, 64 rows per block
    dim3 ggrd((N + 63) / 64);
    if (C == 128)
      gat_gemm_wmma<128><<<ggrd, gblk, 0, stream>>>(X, wpack, h, N, K);
    else
      gat_gemm_wmma<64><<<ggrd, gblk, 0, stream>>>(X, wpack, h, N, K);

    gat_attn_coef<<<(N + 255) / 256, 256, 0, stream>>>(h, avs, avd, es, ed, N, C);

    long tot = (long)N * C;
    gat_init<<<(int)((tot + 255) / 256), 256, 0, stream>>>(mord, denom, agg, N, tot);

    gat_edge_logit_max<<<(ET + 255) / 256, 256, 0, stream>>>(ei, E, N, es, ed, elog, mord);
    gat_edge_exp_sum<<<(ET + 255) / 256, 256, 0, stream>>>(ei, E, N, elog, mord, denom);
    gat_edge_aggregate<<<(ET + 7) / 8, 256, 0, stream>>>(ei, E, N, C / 4, elog, denom, h, agg);

    gat_bias_act<<<(int)((tot + 255) / 256), 256, 0, stream>>>(
        agg, bv, O, tot, C - 1, relu ? 1 : 0);
  };

  layer(x,   W0, as0, ad0, b0, CIN,  CHID, act, true);
  layer(act, W1, as1, ad1, b1, CHID, CHID, act, true);
  layer(act, W2, as2, ad2, b2, CHID, COUT, (float*)d_out, false);
}